// MoAT_53266184405366
// MI455X (gfx1250) — compile-verified
//
#include <hip/hip_runtime.h>
#include <hip/hip_bf16.h>
#include <math.h>

typedef __attribute__((ext_vector_type(2))) float v2f;
typedef __attribute__((ext_vector_type(8))) float v8f;

#define LDS_STRIDE 130   // 130 mod 64 = 2 -> 32-lane column walks hit 32 distinct banks

// ---------------------------------------------------------------------------
// helpers
// ---------------------------------------------------------------------------
__device__ __forceinline__ float sigmoidf_(float x) { return 1.0f / (1.0f + expf(-x)); }

__device__ __forceinline__ void softmax4(const float* __restrict__ src, float* dst) {
  float a0 = src[0], a1 = src[1], a2 = src[2], a3 = src[3];
  float m  = fmaxf(fmaxf(a0, a1), fmaxf(a2, a3));
  float e0 = expf(a0 - m), e1 = expf(a1 - m), e2 = expf(a2 - m), e3 = expf(a3 - m);
  float s  = e0 + e1 + e2 + e3;
  dst[0] = e0 / s; dst[1] = e1 / s; dst[2] = e2 / s; dst[3] = e3 / s;
}

// 256-thread block sum reduction (wave32), result valid on tid 0
__device__ __forceinline__ float block_reduce_sum(float v, float* red8, int tid) {
  #pragma unroll
  for (int off = 16; off > 0; off >>= 1) v += __shfl_down(v, off, 32);
  if ((tid & 31) == 0) red8[tid >> 5] = v;
  __syncthreads();
  float t = 0.0f;
  if (tid == 0) {
    #pragma unroll
    for (int w = 0; w < 8; ++w) t += red8[w];
  }
  return t;
}

// ---------------------------------------------------------------------------
// In-LDS LU (no pivoting; matrix is strictly diagonally dominant) on a 128x128
// system. Blocked right-looking, panel width 16; trailing rank-16 updates done
// with V_WMMA_F32_16X16X4_F32 (4 chained K=4 WMMAs per 16x16 tile).
// Returns per-thread partial sum of log(diag(U)).
// __forceinline__ so InferAddressSpaces sees the __shared__ origin and strips
// generic-pointer null checks from the LDS addressing in the hot loops.
// ---------------------------------------------------------------------------
__device__ __forceinline__ float lu_logdet_128(float (*M)[LDS_STRIDE], int tid) {
  const int wave    = tid >> 5;
  const int lane    = tid & 31;
  const int mrow    = lane & 15;
  const int halfsel = (lane < 16) ? 0 : 2;  // ISA A/B layout: lanes 16-31 hold K=2,3

  for (int k0 = 0; k0 < 128; k0 += 16) {
    // ---- panel factorization: columns k0..k0+15, full remaining height
    for (int j = k0; j < k0 + 16; ++j) {
      __syncthreads();
      const float invp = 1.0f / M[j][j];
      for (int i = j + 1 + tid; i < 128; i += 256) {
        const float m = M[i][j] * invp;
        M[i][j] = m;
        for (int c = j + 1; c < k0 + 16; ++c) M[i][c] -= m * M[j][c];
      }
    }
    __syncthreads();

    const int m0 = k0 + 16;
    if (m0 < 128) {
      // ---- U12 = L11^{-1} * A12 : per-column forward substitution (unit lower L11)
      for (int c = m0 + tid; c < 128; c += 256) {
        float xv[16];
        #pragma unroll
        for (int t = 0; t < 16; ++t) xv[t] = M[k0 + t][c];
        #pragma unroll
        for (int rr = 1; rr < 16; ++rr) {
          float acc = xv[rr];
          for (int t = 0; t < rr; ++t) acc -= M[k0 + rr][k0 + t] * xv[t];
          xv[rr] = acc;
        }
        #pragma unroll
        for (int t = 0; t < 16; ++t) M[k0 + t][c] = xv[t];
      }
      __syncthreads();

      // ---- trailing update A22 -= L21 * U12 via WMMA f32 16x16x4, 8 waves on tiles
      const int nt     = (128 - m0) >> 4;
      const int ntiles = nt * nt;
      for (int t = wave; t < ntiles; t += 8) {
        const int r0 = m0 + (t / nt) * 16;
        const int c0 = m0 + (t % nt) * 16;
        v8f acc = {};
        #pragma unroll
        for (int kk = 0; kk < 4; ++kk) {
          const int kb = k0 + kk * 4 + halfsel;
          // A 16x4: lane -> M row, VGPR0/1 -> K, K+1. (row*130 + kb) is even
          // -> 8B aligned -> single 64-bit LDS load.
          v2f a = *(const v2f*)&M[r0 + mrow][kb];
          v2f b;
          b.x = M[kb + 0][c0 + mrow];     // B 4x16: lane -> N col, VGPR0/1 -> K, K+1
          b.y = M[kb + 1][c0 + mrow];
          acc = __builtin_amdgcn_wmma_f32_16x16x4_f32(
              /*neg_a=*/false, a, /*neg_b=*/false, b,
              /*c_mod=*/(short)0, acc, /*reuse_a=*/false, /*reuse_b=*/false);
        }
        // C/D layout: VGPR r -> rows r (lanes 0-15) / r+8 (lanes 16-31), N = lane%16
        const int crow = r0 + ((lane < 16) ? 0 : 8);
        const int ccol = c0 + mrow;
        #pragma unroll
        for (int r = 0; r < 8; ++r) M[crow + r][ccol] -= acc[r];
      }
    }
    __syncthreads();
  }

  float part = 0.0f;
  for (int j = tid; j < 128; j += 256) part += logf(M[j][j]);
  return part;
}

// Shared pipeline: M initially holds WP shifted so that row i (0..126) is node
// r=i+1, slot j<127 is node c=j+1, slot 127 is node c=0. Converts in place to
// L[1:,1:] + 1e-7 padded with an identity row/col 127, then runs LU.
__device__ __forceinline__ float laplacian_logdet(float (*M)[LDS_STRIDE], float* rs, int tid) {
  __syncthreads();
  if (tid < 127) {
    float s = 0.0f;
    for (int j = 0; j < 128; ++j) s += M[tid][j];   // full row sum of WP (all c != r)
    rs[tid] = s;
  }
  __syncthreads();
  for (int s = tid; s < 128 * 128; s += 256) {
    const int i = s >> 7;
    const int j = s & 127;
    float v;
    if (i == 127)       v = (j == 127) ? 1.0f : 0.0f;   // identity padding
    else if (j == 127)  v = 0.0f;
    else { v = 1e-7f - M[i][j]; if (i == j) v += rs[i]; }
    M[i][j] = v;
  }
  __syncthreads();
  return lu_logdet_128(M, tid);
}

// ---------------------------------------------------------------------------
// Kernel A: prep — V softmax, per-(i,j) 4x4 Sinkhorn on E, symmetric Ws
// One thread per (i,j) pair: 16384 threads = 64 blocks x 256
// ---------------------------------------------------------------------------
__global__ __launch_bounds__(256) void moat_prep_kernel(
    const float* __restrict__ W, const float* __restrict__ Vc,
    const float* __restrict__ Ec, float* __restrict__ Vout,
    float* __restrict__ Eout, float* __restrict__ WsOut) {
  const int p = blockIdx.x * 256 + threadIdx.x;   // 0..16383
  const int i = p >> 7, j = p & 127;

  float vi[4], vj[4];
  softmax4(Vc + i * 4, vi);
  softmax4(Vc + j * 4, vj);
  if (i == 0) {
    #pragma unroll
    for (int k = 0; k < 4; ++k) Vout[j * 4 + k] = vj[k];
  }

  // E block: exp, normalize, 2 Sinkhorn iterations (row->col->total), mask, clip
  float e[16];
  {
    const float* src = Ec + p * 16;
    float s = 0.0f;
    #pragma unroll
    for (int k = 0; k < 16; ++k) { e[k] = expf(src[k]); s += e[k]; }
    float inv = 1.0f / s;
    #pragma unroll
    for (int k = 0; k < 16; ++k) e[k] *= inv;
    #pragma unroll
    for (int it = 0; it < 2; ++it) {
      #pragma unroll
      for (int a = 0; a < 4; ++a) {
        float r = e[a*4+0] + e[a*4+1] + e[a*4+2] + e[a*4+3];
        float f = vi[a] / r;
        e[a*4+0] *= f; e[a*4+1] *= f; e[a*4+2] *= f; e[a*4+3] *= f;
      }
      #pragma unroll
      for (int b = 0; b < 4; ++b) {
        float c = e[b] + e[4+b] + e[8+b] + e[12+b];
        float f = vj[b] / c;
        e[b] *= f; e[4+b] *= f; e[8+b] *= f; e[12+b] *= f;
      }
      float tot = 0.0f;
      #pragma unroll
      for (int k = 0; k < 16; ++k) tot += e[k];
      float it2 = 1.0f / tot;
      #pragma unroll
      for (int k = 0; k < 16; ++k) e[k] *= it2;
    }
    if (i == j) {
      #pragma unroll
      for (int k = 0; k < 16; ++k) e[k] = 0.0f;
    } else {
      #pragma unroll
      for (int k = 0; k < 16; ++k) e[k] = fminf(fmaxf(e[k], 0.0f), 1.0f);
    }
    #pragma unroll
    for (int k = 0; k < 16; ++k) Eout[p * 16 + k] = e[k];
  }

  // Ws = tril(sigmoid(W), -1); Ws = Ws.T + Ws
  float w = 0.0f;
  if (i > j)      w = sigmoidf_(W[i * 128 + j]);
  else if (i < j) w = sigmoidf_(W[j * 128 + i]);
  WsOut[p] = w;
}

// ---------------------------------------------------------------------------
// Kernel B: logdet of L0[1:,1:] + 1e-7 (one block)
// ---------------------------------------------------------------------------
__global__ __launch_bounds__(256) void moat_l0_kernel(
    const float* __restrict__ Ws, float* __restrict__ logdetL0) {
  __shared__ float M[128][LDS_STRIDE];
  __shared__ float rs[128];
  __shared__ float red8[8];
  const int tid = threadIdx.x;

  for (int s = tid; s < 127 * 128; s += 256) {
    const int i  = s >> 7;
    const int js = s & 127;
    const int r  = i + 1;
    const int c  = (js < 127) ? (js + 1) : 0;
    M[i][js] = Ws[r * 128 + c];          // WP == Ws for the base Laplacian
  }
  float part = laplacian_logdet(M, rs, tid);
  float tot  = block_reduce_sum(part, red8, tid);
  if (tid == 0) logdetL0[0] = tot;
}

// ---------------------------------------------------------------------------
// Kernel C: main — one workgroup per batch item
// ---------------------------------------------------------------------------
__global__ __launch_bounds__(256) void moat_main_kernel(
    const int* __restrict__ x, const float* __restrict__ V,
    const float* __restrict__ E, const float* __restrict__ Ws,
    const float* __restrict__ logdetL0, float* __restrict__ y) {
  __shared__ float M[128][LDS_STRIDE];
  __shared__ int   xb[128];
  __shared__ float pr[128];
  __shared__ float rs[128];
  __shared__ float red8[8];
  const int b   = blockIdx.x;
  const int tid = threadIdx.x;

  if (tid < 128) {
    const int xv = x[b * 128 + tid];
    xb[tid] = xv;
    pr[tid] = V[tid * 4 + xv];
  }
  __syncthreads();

  float part = 0.0f;
  if (tid < 128) part = logf(pr[tid] + 1e-7f);   // sum log(Pr + eps)

  // WP[r][c] = Ws[r][c] * E[r,c,x_r,x_c] / (Pr_r * Pr_c), stored shifted
  for (int s = tid; s < 127 * 128; s += 256) {
    const int i  = s >> 7;
    const int js = s & 127;
    const int r  = i + 1;
    const int c  = (js < 127) ? (js + 1) : 0;
    float wp = 0.0f;
    if (r != c) {
      const float w  = Ws[r * 128 + c];
      const float ev = E[(((r << 7) + c) << 4) + (xb[r] << 2) + xb[c]];
      wp = w * ev / (pr[r] * pr[c]);
    }
    M[i][js] = wp;
  }

  part += laplacian_logdet(M, rs, tid);
  float tot = block_reduce_sum(part, red8, tid);
  if (tid == 0) y[b] = tot - logdetL0[0];
}

// ---------------------------------------------------------------------------
// launch
// ---------------------------------------------------------------------------
extern "C" void kernel_launch(void* const* d_in, const int* in_sizes, int n_in,
                              void* d_out, int out_size, void* d_ws, size_t ws_size,
                              hipStream_t stream) {
  const int*   x  = (const int*)d_in[0];
  const float* W  = (const float*)d_in[1];
  const float* Vc = (const float*)d_in[2];
  const float* Ec = (const float*)d_in[3];
  float*       y  = (float*)d_out;

  const int batch = in_sizes[0] / 128;            // 4096

  // workspace layout (floats): V[512] | E[262144] | Ws[16384] | logdetL0[1]
  float* ws    = (float*)d_ws;
  float* Vout  = ws;
  float* Eout  = Vout + 128 * 4;
  float* WsOut = Eout + 128 * 128 * 16;
  float* ld0   = WsOut + 128 * 128;

  moat_prep_kernel<<<64, 256, 0, stream>>>(W, Vc, Ec, Vout, Eout, WsOut);
  moat_l0_kernel<<<1, 256, 0, stream>>>(WsOut, ld0);
  moat_main_kernel<<<batch, 256, 0, stream>>>(x, Vout, Eout, WsOut, ld0, y);
}